// StringGNNBackbone_6923487281886
// MI455X (gfx1250) — compile-verified
//
#include <hip/hip_runtime.h>

#define D_DIM 256
#define TILE_ROWS 64
#define XS_STRIDE 260   // 256 + 4 pad: row*260 % 64 == row*4 -> conflict-free ds_load_b64

typedef float v2f __attribute__((ext_vector_type(2)));
typedef float v8f __attribute__((ext_vector_type(8)));

// ---------------------------------------------------------------------------
// Hardware f32 atomic add (global_atomic_add_f32, resolved at L2 on CDNA5).
// ---------------------------------------------------------------------------
__device__ __forceinline__ void atom_add_f32(float* p, float v) {
#if defined(__AMDGCN__)
    unsafeAtomicAdd(p, v);
#else
    atomicAdd(p, v);
#endif
}

// ---------------------------------------------------------------------------
// deg[c] += ew[e]  (self-loop "+1" folded into rsqrt kernel)
// ---------------------------------------------------------------------------
__global__ __launch_bounds__(256) void degree_kernel(
    const int* __restrict__ col, const float* __restrict__ ew,
    float* __restrict__ deg, int E)
{
    int e = blockIdx.x * blockDim.x + threadIdx.x;
    if (e < E) atom_add_f32(&deg[col[e]], ew[e]);
}

__global__ __launch_bounds__(256) void rsqrt_kernel(
    const float* __restrict__ deg, float* __restrict__ dinv, int N)
{
    int i = blockIdx.x * blockDim.x + threadIdx.x;
    if (i < N) dinv[i] = rsqrtf(deg[i] + 1.0f);
}

// ---------------------------------------------------------------------------
// Out[N,256] = X[N,256] @ W[256,256]^T (+ bias if non-null), via
// V_WMMA_F32_16X16X4_F32.
//
// Block = 8 waves. The block's 64-row X stripe is staged once into LDS
// (padded stride 260 -> conflict-free b64 reads), then each wave computes a
// 64x16 output tile: 4 row sub-tiles share every B (W) fragment, A fragments
// come from LDS. grid = (ceil(N/64), 2); wave col tile = (gy*8+wave)*16.
//
// Fragment layouts (ISA 7.12.2, 32-bit):
//   A 16x4 : lane<16 -> {K=k, K=k+1}, lane>=16 -> {K=k+2, K=k+3}, M = lane&15
//   B 4x16 : VGPR0/1 = rows K (split by half-wave), N = lane&15
//   C/D    : VGPR j -> M = j + 8*(lane>>4), N = lane&15
// ---------------------------------------------------------------------------
__global__ __launch_bounds__(256) void gemm_xwT_wmma(
    const float* __restrict__ X, const float* __restrict__ W,
    const float* __restrict__ bias, float* __restrict__ Out, int N)
{
    __shared__ float Xs[TILE_ROWS * XS_STRIDE];   // ~65 KB of the 320 KB WGP LDS

    const int tid  = threadIdx.x;
    const int wave = tid >> 5;
    const int lane = tid & 31;
    const int half = lane >> 4;
    const int lrow = lane & 15;

    const int r0 = blockIdx.x * TILE_ROWS;
    const int c0 = (blockIdx.y * 8 + wave) * 16;

    // ---- cooperative stage of X[r0 : r0+64, :] into LDS (float4, coalesced)
    {
        const int last = N - 1;
#pragma unroll
        for (int i = 0; i < 16; ++i) {
            const int fidx = (i * 256 + tid) * 4;        // flat float index, 0..16383
            const int lr = fidx >> 8;                    // local row
            const int cc = fidx & 255;                   // column
            int gr = r0 + lr; if (gr > last) gr = last;  // clamp tail rows
            *(float4*)(Xs + lr * XS_STRIDE + cc) =
                *(const float4*)(X + (size_t)gr * D_DIM + cc);
        }
    }
    __syncthreads();

    const float bv = bias ? bias[c0 + lrow] : 0.0f;
    v8f acc0 = {bv, bv, bv, bv, bv, bv, bv, bv};
    v8f acc1 = acc0, acc2 = acc0, acc3 = acc0;

    const float* wp  = W  + (size_t)(c0 + lrow) * D_DIM + 2 * half;   // global, L2-hot
    const float* xs0 = Xs + (0 * 16 + lrow) * XS_STRIDE + 2 * half;   // LDS
    const float* xs1 = Xs + (1 * 16 + lrow) * XS_STRIDE + 2 * half;
    const float* xs2 = Xs + (2 * 16 + lrow) * XS_STRIDE + 2 * half;
    const float* xs3 = Xs + (3 * 16 + lrow) * XS_STRIDE + 2 * half;

#pragma unroll 4
    for (int k = 0; k < D_DIM; k += 4) {
        v2f b  = *(const v2f*)(wp + k);     // global_load_b64 (cache-resident W)
        v2f a0 = *(const v2f*)(xs0 + k);    // ds_load_b64, conflict-free
        v2f a1 = *(const v2f*)(xs1 + k);
        v2f a2 = *(const v2f*)(xs2 + k);
        v2f a3 = *(const v2f*)(xs3 + k);
        acc0 = __builtin_amdgcn_wmma_f32_16x16x4_f32(false, a0, false, b, (short)0, acc0, false, false);
        acc1 = __builtin_amdgcn_wmma_f32_16x16x4_f32(false, a1, false, b, (short)0, acc1, false, false);
        acc2 = __builtin_amdgcn_wmma_f32_16x16x4_f32(false, a2, false, b, (short)0, acc2, false, false);
        acc3 = __builtin_amdgcn_wmma_f32_16x16x4_f32(false, a3, false, b, (short)0, acc3, false, false);
    }

    // ---- epilogue: uniform fast path (all 64 rows in-bounds) avoids
    //      per-store exec-mask churn; only the final row-block is guarded.
    const int ccol = c0 + lrow;
    float* outp = Out + (size_t)(r0 + half * 8) * D_DIM + ccol;
    if (r0 + TILE_ROWS <= N) {
#pragma unroll
        for (int j = 0; j < 8; ++j) {
            outp[(size_t)(j +  0) * D_DIM] = acc0[j];
            outp[(size_t)(j + 16) * D_DIM] = acc1[j];
            outp[(size_t)(j + 32) * D_DIM] = acc2[j];
            outp[(size_t)(j + 48) * D_DIM] = acc3[j];
        }
    } else {
#pragma unroll
        for (int j = 0; j < 8; ++j) {
            int r = r0 + half * 8 + j;
            if (r < N) Out[(size_t)r * D_DIM + ccol] = acc0[j];
            r += 16;
            if (r < N) Out[(size_t)r * D_DIM + ccol] = acc1[j];
            r += 16;
            if (r < N) Out[(size_t)r * D_DIM + ccol] = acc2[j];
            r += 16;
            if (r < N) Out[(size_t)r * D_DIM + ccol] = acc3[j];
        }
    }
}

// ---------------------------------------------------------------------------
// Edge scatter: AGG[col] += H[row] * (dinv[row]*ew*dinv[col]).
// 64 lanes per edge, float4 per lane (256 floats/row). H & AGG are L2-resident
// (19.3 MB each << 192 MB L2), so gathers + atomics stay on-chip.
// ---------------------------------------------------------------------------
__global__ __launch_bounds__(256) void scatter_kernel(
    const float* __restrict__ H, const int* __restrict__ row,
    const int* __restrict__ col, const float* __restrict__ ew,
    const float* __restrict__ dinv, float* __restrict__ AGG, int E)
{
    const int grp  = threadIdx.x >> 6;   // 4 edges per block
    const int lane = threadIdx.x & 63;
    const long e = (long)blockIdx.x * 4 + grp;
    if (e >= E) return;

    const int r = row[e];
    const int c = col[e];
    const float norm = dinv[r] * ew[e] * dinv[c];   // broadcast loads within group

    const float4 hv = *(const float4*)(H + (size_t)r * D_DIM + lane * 4);
    float* dst = AGG + (size_t)c * D_DIM + lane * 4;
    atom_add_f32(dst + 0, hv.x * norm);
    atom_add_f32(dst + 1, hv.y * norm);
    atom_add_f32(dst + 2, hv.z * norm);
    atom_add_f32(dst + 3, hv.w * norm);
}

// ---------------------------------------------------------------------------
// out = agg + h * dinv[i]^2 + bias + residual   (float4 vectorized)
// ---------------------------------------------------------------------------
__global__ __launch_bounds__(256) void combine_kernel(
    const float* __restrict__ agg, const float* __restrict__ h,
    const float* __restrict__ dinv, const float* __restrict__ bias,
    const float* __restrict__ res, float* __restrict__ out, int N)
{
    const long idx = ((long)blockIdx.x * blockDim.x + threadIdx.x) * 4;
    if (idx >= (long)N * D_DIM) return;
    const int i = (int)(idx >> 8);
    const int d = (int)(idx & (D_DIM - 1));
    float s = dinv[i];
    s = s * s;
    const float4 a  = *(const float4*)(agg + idx);
    const float4 hv = *(const float4*)(h + idx);
    const float4 bv = *(const float4*)(bias + d);
    const float4 rv = *(const float4*)(res + idx);
    float4 o;
    o.x = a.x + hv.x * s + bv.x + rv.x;
    o.y = a.y + hv.y * s + bv.y + rv.y;
    o.z = a.z + hv.z * s + bv.z + rv.z;
    o.w = a.w + hv.w * s + bv.w + rv.w;
    *(float4*)(out + idx) = o;
}

// ---------------------------------------------------------------------------
// Host-side orchestration (graph-capture safe: only async memsets + launches)
// ---------------------------------------------------------------------------
extern "C" void kernel_launch(void* const* d_in, const int* in_sizes, int n_in,
                              void* d_out, int out_size, void* d_ws, size_t ws_size,
                              hipStream_t stream) {
    const float* x0 = (const float*)d_in[0];   // [N, 256]
    const int*   ei = (const int*)d_in[1];     // [2, E] (int32)
    const float* ew = (const float*)d_in[2];   // [E]
    const float* W1 = (const float*)d_in[3];
    const float* b1 = (const float*)d_in[4];
    const float* W2 = (const float*)d_in[5];
    const float* b2 = (const float*)d_in[6];
    const float* Wp = (const float*)d_in[7];
    const float* bp = (const float*)d_in[8];

    const int N = in_sizes[0] / D_DIM;         // 18870
    const int E = in_sizes[2];                 // 1,200,000
    const int* row = ei;
    const int* col = ei + E;

    const size_t Npad  = ((size_t)N + 63) & ~(size_t)63;
    const size_t elems = (size_t)N * D_DIM;

    float* deg  = (float*)d_ws;        // Npad
    float* dinv = deg + Npad;          // Npad
    float* H    = dinv + Npad;         // N*D
    float* AGG  = H + elems;           // N*D
    float* XC   = AGG + elems;         // N*D

    // --- degrees & symmetric normalization -------------------------------
    hipMemsetAsync(deg, 0, Npad * sizeof(float), stream);
    degree_kernel<<<(E + 255) / 256, 256, 0, stream>>>(col, ew, deg, E);
    rsqrt_kernel<<<(N + 255) / 256, 256, 0, stream>>>(deg, dinv, N);

    const dim3 gg((N + 63) / 64, 2);
    const int scat_grid = (E + 3) / 4;
    const int comb_grid = (int)((elems / 4 + 255) / 256);

    // --- GCNConv layer 1 + residual --------------------------------------
    gemm_xwT_wmma<<<gg, 256, 0, stream>>>(x0, W1, nullptr, H, N);
    hipMemsetAsync(AGG, 0, elems * sizeof(float), stream);
    scatter_kernel<<<scat_grid, 256, 0, stream>>>(H, row, col, ew, dinv, AGG, E);
    combine_kernel<<<comb_grid, 256, 0, stream>>>(AGG, H, dinv, b1, x0, XC, N);

    // --- GCNConv layer 2 + residual (in-place into XC) -------------------
    gemm_xwT_wmma<<<gg, 256, 0, stream>>>(XC, W2, nullptr, H, N);
    hipMemsetAsync(AGG, 0, elems * sizeof(float), stream);
    scatter_kernel<<<scat_grid, 256, 0, stream>>>(H, row, col, ew, dinv, AGG, E);
    combine_kernel<<<comb_grid, 256, 0, stream>>>(AGG, H, dinv, b2, XC, XC, N);

    // --- post_mp projection: out = XC @ Wp^T + bp ------------------------
    gemm_xwT_wmma<<<gg, 256, 0, stream>>>(XC, Wp, bp, (float*)d_out, N);
}